// SNNBinaryClassifier_65257733095648
// MI455X (gfx1250) — compile-verified
//
#include <hip/hip_runtime.h>

typedef __attribute__((ext_vector_type(2))) float v2f;
typedef __attribute__((ext_vector_type(8))) float v8f;

#define DECAY_F  0.9f
#define THRESH_F 1.0f

// ---------------------------------------------------------------------------
// Kernel 1: h1 = x @ W1^T + b1 using V_WMMA_F32_16X16X4_F32 (exact fp32).
// One wave (32 lanes) computes one 16x16 output tile; K-loop of D/4 WMMAs.
//
// A (16x4, MxK) layout: lane L holds M = L%16, K = kbase + 2*(L/16) + {0,1}
// B (4x16, KxN) layout: lane L holds N = L%16, K = kbase + 2*(L/16) + {0,1}
// Both map to contiguous float2 loads since x[B,D] and W1[H,D] are K-inner.
// C/D (16x16 f32): VGPR r, lanes 0-15 -> M=r, N=lane; lanes 16-31 -> M=8+r.
// ---------------------------------------------------------------------------
__global__ __launch_bounds__(256) void snn_fc1_wmma(
    const float* __restrict__ x,   // [B, D]
    const float* __restrict__ W1,  // [H, D]
    const float* __restrict__ b1,  // [H]
    float* __restrict__ h1,        // [B, H]  (workspace)
    int D, int H)
{
  const int lane   = threadIdx.x & 31;
  const int wave   = threadIdx.x >> 5;
  const int tilesN = H >> 4;
  const int tile   = blockIdx.x * 8 + wave;   // grid sized exactly: no tail
  const int row0   = (tile / tilesN) << 4;    // tile over B
  const int col0   = (tile % tilesN) << 4;    // tile over H

  const int r16  = lane & 15;
  const int koff = (lane >> 4) << 1;          // 0 for lanes 0-15, 2 for 16-31

  const float* xrow = x  + (size_t)(row0 + r16) * D;   // A row (M = r16)
  const float* wrow = W1 + (size_t)(col0 + r16) * D;   // B col (N = r16)

  v8f acc = {};
  for (int k = 0; k < D; k += 4) {
    v2f a = *(const v2f*)(xrow + k + koff);
    v2f b = *(const v2f*)(wrow + k + koff);
    // 8 args: (neg_a, A, neg_b, B, c_mod, C, reuse_a, reuse_b)
    acc = __builtin_amdgcn_wmma_f32_16x16x4_f32(
        false, a, false, b, (short)0, acc, false, false);
  }

  const float bias = b1[col0 + r16];          // same N for all 8 acc regs
  const int   mhi  = (lane >> 4) << 3;        // M offset 0 or 8
  #pragma unroll
  for (int r = 0; r < 8; ++r) {
    h1[(size_t)(row0 + mhi + r) * H + (col0 + r16)] = acc[r] + bias;
  }
}

// ---------------------------------------------------------------------------
// Kernel 2: LIF time loop. One wave32 per batch row; each lane owns 16
// contiguous H elements (H=512). mem state lives in VGPRs for all T steps.
// Per step: fused decay+fire+reset, two masked partial sums, wave32 butterfly
// reduction, sign-count accumulate. Output = spike counts / T.
// ---------------------------------------------------------------------------
__global__ __launch_bounds__(256) void snn_time_loop(
    const float* __restrict__ h1,   // [B, H]
    const float* __restrict__ W2,   // [2, H]
    const float* __restrict__ b2,   // [2]
    const int*   __restrict__ tsp,  // time_steps (device scalar)
    float* __restrict__ out,        // [B, 2]
    int H)
{
  const int lane = threadIdx.x & 31;
  const int wave = threadIdx.x >> 5;
  const int row  = blockIdx.x * 8 + wave;     // grid sized exactly: no tail
  const int T    = *tsp;

  const int hbase = lane * 16;                // H/32 = 16 elements per lane

  float h1v[16], w0v[16], w1v[16], mem[16];
  {
    const float4* hp = (const float4*)(h1 + (size_t)row * H + hbase);
    const float4* w0 = (const float4*)(W2 + hbase);
    const float4* w1 = (const float4*)(W2 + H + hbase);
    #pragma unroll
    for (int q = 0; q < 4; ++q) {
      float4 a = hp[q], b = w0[q], c = w1[q];
      h1v[4*q+0] = a.x; h1v[4*q+1] = a.y; h1v[4*q+2] = a.z; h1v[4*q+3] = a.w;
      w0v[4*q+0] = b.x; w0v[4*q+1] = b.y; w0v[4*q+2] = b.z; w0v[4*q+3] = b.w;
      w1v[4*q+0] = c.x; w1v[4*q+1] = c.y; w1v[4*q+2] = c.z; w1v[4*q+3] = c.w;
    }
  }
  #pragma unroll
  for (int j = 0; j < 16; ++j) mem[j] = 0.0f;

  const float bb0 = b2[0], bb1 = b2[1];
  float cnt0 = 0.0f, cnt1 = 0.0f;

  for (int t = 0; t < T; ++t) {
    float p0 = 0.0f, p1 = 0.0f;
    #pragma unroll
    for (int j = 0; j < 16; ++j) {
      float m    = __builtin_fmaf(DECAY_F, mem[j], h1v[j]);  // leaky integrate
      float fire = (m > THRESH_F) ? 1.0f : 0.0f;             // spike
      mem[j]     = m - fire * THRESH_F;                      // soft reset
      p0 = __builtin_fmaf(fire, w0v[j], p0);                 // masked dot W2[0]
      p1 = __builtin_fmaf(fire, w1v[j], p1);                 // masked dot W2[1]
    }
    // wave32 butterfly reduction; all lanes end with the full sum
    #pragma unroll
    for (int msk = 16; msk >= 1; msk >>= 1) {
      p0 += __shfl_xor(p0, msk, 32);
      p1 += __shfl_xor(p1, msk, 32);
    }
    cnt0 += ((p0 + bb0) > 0.0f) ? 1.0f : 0.0f;
    cnt1 += ((p1 + bb1) > 0.0f) ? 1.0f : 0.0f;
  }

  if (lane == 0) {
    float invT = 1.0f / (float)T;
    out[row * 2 + 0] = cnt0 * invT;
    out[row * 2 + 1] = cnt1 * invT;
  }
}

// ---------------------------------------------------------------------------
// inputs: x[B,D], W1[H,D], b1[H], W2[2,H], b2[2], time_steps (int scalar)
// output: [B, 2] float32
// ---------------------------------------------------------------------------
extern "C" void kernel_launch(void* const* d_in, const int* in_sizes, int n_in,
                              void* d_out, int out_size, void* d_ws, size_t ws_size,
                              hipStream_t stream) {
  const float* x  = (const float*)d_in[0];
  const float* W1 = (const float*)d_in[1];
  const float* b1 = (const float*)d_in[2];
  const float* W2 = (const float*)d_in[3];
  const float* b2 = (const float*)d_in[4];
  const int*   ts = (const int*)d_in[5];

  const int H = in_sizes[2];            // 512
  const int D = in_sizes[1] / H;        // 256
  const int B = in_sizes[0] / D;        // 4096

  float* h1 = (float*)d_ws;             // B*H*4 = 8 MB scratch

  // Kernel 1: one wave per 16x16 tile, 8 waves per block.
  const int tiles = (B >> 4) * (H >> 4);          // 8192
  snn_fc1_wmma<<<tiles / 8, 256, 0, stream>>>(x, W1, b1, h1, D, H);

  // Kernel 2: one wave per row, 8 rows per block.
  snn_time_loop<<<B / 8, 256, 0, stream>>>(h1, W2, b2, ts, (float*)d_out, H);
}